// SlowROIPool_43705587204143
// MI455X (gfx1250) — compile-verified
//
#include <hip/hip_runtime.h>
#include <cstdint>
#include <cstddef>

// Adaptive ROI max-pool (SlowROIPool), MI455X/gfx1250.
// Grid = (R, C/CCH); one workgroup per (ROI, 32-channel chunk). Wave 0 issues
// a single Tensor Data Mover descriptor that DMAs the 3-D crop tile
// (cropW x cropH x 32ch, strides W and H*W) from global memory into LDS,
// waits on TENSORcnt, then all 8 wave32s compute the 7x7 adaptive-max bins
// from LDS. images (10.2 MB) is L2-resident (192 MB L2), so this is
// latency-bound; the TDM turns ~450 scattered per-lane VMEM loads into one
// async DMA per block, and the 2-D grid removes the runtime div preamble.

#define POOL_S 7
#define CCH    32          // channels staged per block
#define MAXD   22          // crop dim bound: ceil(0.4*50)+1 = 21, +1 slack

typedef __attribute__((ext_vector_type(4))) unsigned int v4u;
typedef __attribute__((ext_vector_type(8))) int          v8i;
typedef __attribute__((ext_vector_type(4))) int          v4i;

__global__ __launch_bounds__(256)
void roi_pool_tdm_kernel(const float* __restrict__ images,
                         const float4* __restrict__ rois,
                         const int*   __restrict__ roi_idx,
                         float*       __restrict__ out,
                         int C, int H, int W)
{
    __shared__ float smem[CCH * MAXD * MAXD];   // 61,952 B

    const int r  = (int)blockIdx.x;             // ROI
    const int c0 = (int)blockIdx.y * CCH;       // channel chunk base

    // ---- uniform ROI geometry (match jnp.floor/ceil in f32) ----
    const float4 box = rois[r];                 // one s_load_b128
    int x1 = (int)floorf(box.x * (float)W);
    int y1 = (int)floorf(box.y * (float)H);
    int x2 = (int)ceilf (box.z * (float)W);
    int y2 = (int)ceilf (box.w * (float)H);
    int n  = roi_idx[r];

    // force into SGPRs: the TDM descriptor must be scalar
    x1 = __builtin_amdgcn_readfirstlane(x1);
    y1 = __builtin_amdgcn_readfirstlane(y1);
    x2 = __builtin_amdgcn_readfirstlane(x2);
    y2 = __builtin_amdgcn_readfirstlane(y2);
    n  = __builtin_amdgcn_readfirstlane(n);

    int cW = x2 - x1; if (cW > MAXD) cW = MAXD; if (cW < 1) cW = 1;
    int cH = y2 - y1; if (cH > MAXD) cH = MAXD; if (cH < 1) cH = 1;

    // ---- wave 0: one TDM DMA of the whole (cW x cH x CCH) tile ----
    if (threadIdx.x < 32) {
        const uint64_t gaddr = (uint64_t)(uintptr_t)(images
            + ((((size_t)n * (size_t)C + (size_t)c0) * (size_t)H + (size_t)y1)
               * (size_t)W + (size_t)x1));
        // generic LDS pointer: low 32 bits == LDS byte offset (ISA 10.2)
        const uint32_t laddr = (uint32_t)(uintptr_t)&smem[0];

        v4u g0 = {};
        g0[0] = 1u;                                  // count=1 (valid), user mode
        g0[1] = laddr;                               // lds_addr
        g0[2] = (uint32_t)gaddr;                     // global_addr[31:0]
        g0[3] = ((uint32_t)(gaddr >> 32) & 0x01FFFFFFu) | (2u << 30); // type=2

        const uint32_t s0 = (uint32_t)W;             // tensor_dim0_stride
        const uint32_t s1 = (uint32_t)(H * W);       // tensor_dim1_stride
        v8i g1 = {};
        g1[0] = (int)(2u << 16);                     // data_size=2 -> 4 bytes
        g1[1] = (int)((uint32_t)cW << 16);           // tensor_dim0[15:0]
        g1[2] = (int)((uint32_t)cH << 16);           // tensor_dim1[15:0]
        g1[3] = (int)((uint32_t)cW << 16);           // tile_dim0
        g1[4] = (int)(((uint32_t)CCH << 16) | (uint32_t)cH); // tile_dim1|tile_dim2
        g1[5] = (int)s0;                             // stride0[31:0]
        g1[6] = (int)(s1 << 16);                     // stride0[47:32]=0 | stride1[15:0]
        g1[7] = (int)(s1 >> 16);                     // stride1[47:16]

        v4i g2 = {};
        g2[0] = CCH;                                 // tensor_dim2
        g2[1] = 1;                                   // tensor_dim3 (unused)
        g2[2] = 0;                                   // tensor_dim2_stride lo
        g2[3] = 0;                                   // stride hi | tile_dim3=0
        v4i g3 = {};                                 // dims 4+ unused
        v8i g4 = {};                                 // extension slot: all-disabled

        __builtin_amdgcn_tensor_load_to_lds(g0, g1, g2, g3, g4, 0 /*cpol*/);
        __builtin_amdgcn_s_wait_tensorcnt(0);
    }
    __syncthreads();

    // ---- compute: 32 channels x 49 bins from LDS ----
    // thread t: channel = t>>3, bins b = (t&7), (t&7)+8, ... (< 49)
    const int c   = (int)threadIdx.x >> 3;
    const int b0  = (int)threadIdx.x & 7;
    const float* base = &smem[c * cH * cW];
    float* outc = out + (((size_t)r * (size_t)C + (size_t)(c0 + c))
                         * (POOL_S * POOL_S));

    for (int b = b0; b < POOL_S * POOL_S; b += 8) {
        const int i = b / POOL_S;
        const int j = b - i * POOL_S;
        // PyTorch adaptive bins, crop-local: [floor(i*s/S), ceil((i+1)*s/S))
        const int sy = (i * cH) / POOL_S;
        const int ey = ((i + 1) * cH + POOL_S - 1) / POOL_S;
        const int sx = (j * cW) / POOL_S;
        const int ex = ((j + 1) * cW + POOL_S - 1) / POOL_S;

        float m = -__builtin_huge_valf();
        for (int y = sy; y < ey; ++y) {
            const float* row = base + y * cW;
            for (int x = sx; x < ex; ++x)
                m = fmaxf(m, row[x]);
        }
        outc[b] = m;
    }
}

extern "C" void kernel_launch(void* const* d_in, const int* in_sizes, int n_in,
                              void* d_out, int out_size, void* d_ws, size_t ws_size,
                              hipStream_t stream) {
    const float*  images  = (const float*)d_in[0];
    const float4* rois    = (const float4*)d_in[1];
    const int*    roi_idx = (const int*)d_in[2];
    float*        out     = (float*)d_out;

    const int R = in_sizes[2];        // 256
    const int C = 256, H = 50, W = 50;

    const dim3 grid((unsigned)R, (unsigned)(C / CCH));
    const dim3 block(256);
    roi_pool_tdm_kernel<<<grid, block, 0, stream>>>(images, rois, roi_idx, out,
                                                    C, H, W);
}